// GuidedLocalConv2d_75007308857584
// MI455X (gfx1250) — compile-verified
//
#include <hip/hip_runtime.h>
#include <stdint.h>

// GuidedLocalConv2d on MI455X (gfx1250, wave32).
// out[b,o,h,w] = sum_{c,k} patch[b,c,k,h,w] * W[b,c,k,o,h,w]
// Memory-bound: 302MB weight stream / 23.3 TB/s ~= 13us floor.
//
// 512-thr WG = 16 waves = 16 consecutive pixels. Weight chunks are streamed
// memory->LDS with GLOBAL_LOAD_ASYNC_TO_LDS_B128 (ASYNCcnt, no VGPR staging),
// double-buffered, then each wave ds-reads its pixel's slab in
// V_WMMA_F32_16X16X4_F32 A-layout (M=o, K=ck). B operand = patch values
// broadcast across N; patches precomputed once per wave into LDS.

typedef float v2f __attribute__((ext_vector_type(2)));
typedef float v8f __attribute__((ext_vector_type(8)));

#define CI 32
#define CO 32
#define CK 288                    // Ci * 9 reduction length
#define OSTRIDE 4096              // H*W
#define CKSTRIDE (CO * OSTRIDE)   // 131072 elems between (c,k) slices

#define NPX 16                    // pixels per workgroup (= waves per WG)
#define PXP 20                    // padded px dim: 80B rows (16B-aligned b128)
#define CKC 8                     // ck slices per chunk
#define NCHUNK (CK / CKC)         // 36
#define WBUF_HALF (CKC * CO * PXP * 4)            // 20480 B per buffer
#define CHUNK_GB ((uint32_t)(CKC * CKSTRIDE) * 4u) // 4 MiB global stride/chunk

__global__ __launch_bounds__(512)
void glc2d_wmma_async_kernel(const float* __restrict__ input,
                             const float* __restrict__ weights,
                             float* __restrict__ out)
{
    __shared__ __attribute__((aligned(16))) float wbuf[2][CKC][CO][PXP]; // 40960 B
    __shared__ __attribute__((aligned(16))) float pbuf[NPX][CK + 2];     // 18560 B

    const int t    = threadIdx.x;
    const int lane = t & 31;
    const int wave = t >> 5;                 // local pixel
    const int px0  = blockIdx.x * NPX;
    const int b    = px0 >> 12;
    const int hw0  = px0 & 4095;
    const int h    = hw0 >> 6;
    const int w0   = hw0 & 63;
    const int w    = w0 + wave;

    // ---- per-wave patch preload: p[ck], ck = 0..287 ----
    const float* inB = input + (size_t)b * (CI * OSTRIDE);
    for (int tt = 0; tt < 9; ++tt) {
        const int cki = tt * 32 + lane;
        const int c   = cki / 9;
        const int k   = cki - c * 9;
        const int hh  = h + k / 3 - 1;
        const int ww  = w + (k % 3) - 1;
        float p = 0.0f;
        if (hh >= 0 && hh < 64 && ww >= 0 && ww < 64)
            p = inB[c * OSTRIDE + hh * 64 + ww];
        pbuf[wave][cki] = p;                 // same-wave LDS ops are in-order
    }

    // ---- async producer mapping: t -> (px quad, o, e0); 2 x b128 per chunk ----
    const int q  = t & 3;                    // pixel quad: px = 4q..4q+3
    const int po = (t >> 2) & 31;            // o
    const int e0 = t >> 7;                   // 0..3 ; handles ck slices e0, e0+4

    const float* wB = weights + (size_t)b * CK * CKSTRIDE;   // uniform saddr
    uint32_t goffA = ((uint32_t)e0 * CKSTRIDE + (uint32_t)po * OSTRIDE
                    + (uint32_t)(h * 64 + w0) + (uint32_t)q * 4u) * 4u;
    uint32_t goffB = goffA + (uint32_t)(4 * CKSTRIDE) * 4u;

    const uint32_t lA0 = (uint32_t)(uintptr_t)&wbuf[0][e0][po][q * 4];
    const uint32_t lB0 = lA0 + (uint32_t)(4 * CO * PXP * 4);

    // issue one chunk (16KB) of weights: memory -> LDS, tracked by ASYNCcnt
    auto issue_chunk = [&](uint32_t gA, uint32_t gB, int buf) {
        const uint32_t lA = lA0 + (uint32_t)buf * WBUF_HALF;
        const uint32_t lB = lB0 + (uint32_t)buf * WBUF_HALF;
        asm volatile("global_load_async_to_lds_b128 %0, %1, %2\n\t"
                     "global_load_async_to_lds_b128 %3, %4, %2"
                     :: "v"(lA), "v"(gA), "s"(wB), "v"(lB), "v"(gB)
                     : "memory");
    };

    // prologue: chunk 0 into buffer 0
    issue_chunk(goffA, goffB, 0);
    asm volatile("s_wait_asynccnt 0x0" ::: "memory");
    __syncthreads();

    const int mo   = lane & 15;              // A: M = o
    const int koff = (lane >> 4) * 2;        // A/B: K pair 0/1 vs 2/3

    v8f acc_lo = {};                         // o = 0..15
    v8f acc_hi = {};                         // o = 16..31

    for (int chunk = 0; chunk < NCHUNK; ++chunk) {
        const int buf = chunk & 1;

        // stream chunk+1 into the other buffer while we compute on this one
        if (chunk + 1 < NCHUNK) {
            goffA += CHUNK_GB;
            goffB += CHUNK_GB;
            issue_chunk(goffA, goffB, buf ^ 1);
            __builtin_prefetch((const char*)wB + goffA + CHUNK_GB, 0, 1);
        }

        const int ck0 = chunk * CKC;
#pragma unroll
        for (int s = 0; s < CKC / 4; ++s) {
            const int cl = s * 4 + koff;

            // B operand: p[ck], p[ck+1] broadcast across the half-wave
            v2f bmat = *(const v2f*)&pbuf[wave][ck0 + cl];

            // A operands in 16x4 layout: lane l -> (M=l%16, K=koff+j)
            v2f a_lo, a_hi;
            a_lo[0] = wbuf[buf][cl    ][mo     ][wave];
            a_lo[1] = wbuf[buf][cl + 1][mo     ][wave];
            a_hi[0] = wbuf[buf][cl    ][mo + 16][wave];
            a_hi[1] = wbuf[buf][cl + 1][mo + 16][wave];

            acc_lo = __builtin_amdgcn_wmma_f32_16x16x4_f32(
                false, a_lo, false, bmat, (short)0, acc_lo, false, false);
            acc_hi = __builtin_amdgcn_wmma_f32_16x16x4_f32(
                false, a_hi, false, bmat, (short)0, acc_hi, false, false);
        }

        asm volatile("s_wait_asynccnt 0x0" ::: "memory");
        __syncthreads();
    }

    // ---- store: D columns identical; lane 0 -> M=0..7, lane 16 -> M=8..15 ----
    float* outPx = out + (size_t)b * (CO * OSTRIDE) + (size_t)(h * 64 + w);
    if ((lane & 15) == 0) {
        const int half8 = (lane >> 4) * 8;
#pragma unroll
        for (int rr = 0; rr < 8; ++rr) {
            outPx[(size_t)(rr + half8) * OSTRIDE]      = acc_lo[rr];
            outPx[(size_t)(rr + half8 + 16) * OSTRIDE] = acc_hi[rr];
        }
    }
}

extern "C" void kernel_launch(void* const* d_in, const int* in_sizes, int n_in,
                              void* d_out, int out_size, void* d_ws, size_t ws_size,
                              hipStream_t stream) {
    (void)in_sizes; (void)n_in; (void)d_ws; (void)ws_size; (void)out_size;
    const float* input   = (const float*)d_in[0];   // (2,32,64,64)
    const float* weights = (const float*)d_in[1];   // (2,32,9,32,64,64)
    float* out           = (float*)d_out;           // (2,32,64,64)

    const int totalPx = 2 * 64 * 64;                // 8192 pixels
    dim3 grid(totalPx / NPX);                       // 512 WGs
    dim3 block(512);                                // 16 waves, 1 pixel each
    glc2d_wmma_async_kernel<<<grid, block, 0, stream>>>(input, weights, out);
}